// Receiver_3685081940497
// MI455X (gfx1250) — compile-verified
//
#include <hip/hip_runtime.h>
#include <stdint.h>

// ---------------- problem constants (match reference) ----------------
#define NN      100000
#define EE      1000000
#define HID     64
#define EMB     30
#define HEADS   2
#define FEAT    512
#define VOCAB   10
#define GROWS   (NN * HEADS)          // 200000 rows in layer-2 GEMM
#define Z1W     (HEADS * HID)         // 128 floats per node, layer 1
#define Z2W     (HEADS * HEADS * EMB) // 120 floats per node, layer 2

typedef __attribute__((ext_vector_type(2))) float v2f;
typedef __attribute__((ext_vector_type(8))) float v8f;

// gfx1250 async-LDS builtin takes (v4i __device__*, v4i __shared__*, imm, imm)
// where v4i is a GCC-style vector_size(16) int vector (per hipcc diagnostic).
// Integer casts are the legal way to construct addrspace-qualified pointers.
typedef int v4i_vs __attribute__((vector_size(16)));
typedef __attribute__((address_space(1))) v4i_vs gv4i_t;
typedef __attribute__((address_space(3))) v4i_vs lv4i_t;
#define AS_GLOBAL(p) ((gv4i_t*)(uintptr_t)(p))
#define AS_LDS(p)    ((lv4i_t*)(uint32_t)(uintptr_t)(p))

#if defined(__HIP_DEVICE_COMPILE__) && __has_builtin(__builtin_amdgcn_global_load_async_to_lds_b128)
#define HAVE_ASYNC_LDS 1
#else
#define HAVE_ASYNC_LDS 0
#endif

// ---------------- workspace layout (float offsets) ----------------
// z1 aliased with r2 (z1 dead before r2 accumulation starts)
#define OFF_Z1    ((size_t)0)                    // N*128
#define OFF_R2    OFF_Z1                         // N*120 (alias)
#define OFF_R1    ((size_t)12800000)             // N*128
#define OFF_Z2    ((size_t)25600000)             // N*120
#define OFF_EL1   ((size_t)37600000)             // N*2
#define OFF_ER1   ((size_t)37800000)
#define OFF_M1    ((size_t)38000000)
#define OFF_DEN1  ((size_t)38200000)
#define OFF_EL2   ((size_t)38400000)             // N*4
#define OFF_ER2   ((size_t)38800000)
#define OFF_M2    ((size_t)39200000)
#define OFF_DEN2  ((size_t)39600000)
#define OFF_A00   ((size_t)40000000)             // 32
#define OFF_HIMG  ((size_t)40000064)             // 32

__device__ __forceinline__ float lrelu(float x) { return x > 0.f ? x : 0.2f * x; }
__device__ __forceinline__ float sigm(float x)  { return 1.f / (1.f + __expf(-x)); }

// Native CDNA5 float max atomic (no-return, device scope). Avoids CAS retry
// storms on the heavily contended segment-max arrays (2-4M updates onto
// 200-400k addresses); resolves in the L2 atomic units.
__device__ __forceinline__ void atomicMaxF(float* addr, float val) {
    asm volatile("global_atomic_max_num_f32 %0, %1, off scope:SCOPE_DEV"
                 :: "v"((unsigned long long)(uintptr_t)addr), "v"(val)
                 : "memory");
}

// ---------------- K0: dense image path (tiny) ----------------
__global__ void img_kernel(const float* __restrict__ x, const float* __restrict__ W_lin1,
                           const float* __restrict__ w2, const float* __restrict__ w3,
                           float* __restrict__ h_img) {
    __shared__ float hi[EMB];
    int t = threadIdx.x;
    if (t < EMB) {
        float acc = 0.f;
        for (int k = 0; k < FEAT; ++k) acc += W_lin1[t * FEAT + k] * x[k];
        hi[t] = acc;
    }
    __syncthreads();
    if (t < EMB) {
        float acc = 0.f;
        float h = hi[t];
        for (int k = 0; k < HID; ++k) acc += w3[k] * sigm(w2[k] * h);
        h_img[t] = sigm(acc);
    }
}

// ---------------- K1: init layer-1 accumulators + a00 ----------------
__global__ void init1_kernel(float* __restrict__ r1, float* __restrict__ m1,
                             float* __restrict__ den1, float* __restrict__ a00) {
    size_t i = (size_t)blockIdx.x * blockDim.x + threadIdx.x;
    size_t stride = (size_t)gridDim.x * blockDim.x;
    for (size_t k = i; k < (size_t)NN * Z1W; k += stride) r1[k] = 0.f;
    for (size_t k = i; k < (size_t)NN * HEADS; k += stride) { m1[k] = -1e30f; den1[k] = 0.f; }
    if (i < 32) a00[i] = 0.f;
}

// ---------------- K2: z1 = feat@W1, el1/er1 ----------------
__global__ __launch_bounds__(128) void z1_kernel(const float* __restrict__ feat,
                                                 const float* __restrict__ W1,
                                                 const float* __restrict__ al1,
                                                 const float* __restrict__ ar1,
                                                 float* __restrict__ z1,
                                                 float* __restrict__ el1,
                                                 float* __restrict__ er1) {
    int n = blockIdx.x, j = threadIdx.x;
    float f0 = feat[n * 3 + 0], f1 = feat[n * 3 + 1], f2 = feat[n * 3 + 2];
    float z = f0 * W1[j] + f1 * W1[Z1W + j] + f2 * W1[2 * Z1W + j];
    z1[(size_t)n * Z1W + j] = z;
    __shared__ float sl[Z1W], sr[Z1W];
    sl[j] = z * al1[j]; sr[j] = z * ar1[j];
    __syncthreads();
    for (int s = 32; s >= 1; s >>= 1) {
        if ((j & 63) < s) { sl[j] += sl[j + s]; sr[j] += sr[j + s]; }
        __syncthreads();
    }
    if ((j & 63) == 0) { int h = j >> 6; el1[n * 2 + h] = sl[j]; er1[n * 2 + h] = sr[j]; }
}

// ---------------- K3/K4: layer-1 edge softmax (max, denom) ----------------
__global__ void emax1_kernel(const int* __restrict__ src, const int* __restrict__ dst,
                             const float* __restrict__ el, const float* __restrict__ er,
                             float* __restrict__ m) {
    int i = blockIdx.x * blockDim.x + threadIdx.x;
    if (i >= EE) return;
    int s = src[i], d = dst[i];
    for (int h = 0; h < 2; ++h) atomicMaxF(&m[d * 2 + h], lrelu(el[s * 2 + h] + er[d * 2 + h]));
}

__global__ void eden1_kernel(const int* __restrict__ src, const int* __restrict__ dst,
                             const float* __restrict__ el, const float* __restrict__ er,
                             const float* __restrict__ m, float* __restrict__ den) {
    int i = blockIdx.x * blockDim.x + threadIdx.x;
    if (i >= EE) return;
    int s = src[i], d = dst[i];
    for (int h = 0; h < 2; ++h) {
        float e = lrelu(el[s * 2 + h] + er[d * 2 + h]);
        atomicAdd(&den[d * 2 + h], __expf(e - m[d * 2 + h]));
    }
}

// ---------------- K5: layer-1 weighted scatter (wave32 per edge) ----------------
__global__ __launch_bounds__(256) void agg1_kernel(const int* __restrict__ src, const int* __restrict__ dst,
                                                   const float* __restrict__ el, const float* __restrict__ er,
                                                   const float* __restrict__ m, const float* __restrict__ den,
                                                   const float* __restrict__ z1, float* __restrict__ r1) {
    int eid = (blockIdx.x * blockDim.x + threadIdx.x) >> 5;
    int lane = threadIdx.x & 31;
    if (eid >= EE) return;
    int s = src[eid], d = dst[eid];
    const float* zrow = z1 + (size_t)s * Z1W + lane * 4;
    __builtin_prefetch(zrow, 0, 1);            // global_prefetch_b8: start L2 fetch early
    int h = lane >> 4;  // lanes 0-15 -> head 0 (feat 0..63), lanes 16-31 -> head 1
    float e = lrelu(el[s * 2 + h] + er[d * 2 + h]);
    float alpha = __expf(e - m[d * 2 + h]) / den[d * 2 + h];
    const float4 zv = *(const float4*)zrow;
    float* out = r1 + (size_t)d * Z1W + lane * 4;
    atomicAdd(out + 0, alpha * zv.x);
    atomicAdd(out + 1, alpha * zv.y);
    atomicAdd(out + 2, alpha * zv.z);
    atomicAdd(out + 3, alpha * zv.w);
}

// ---------------- K6: r1 = relu(r1 + b1) ----------------
__global__ void bias1_kernel(float* __restrict__ r1, const float* __restrict__ b1) {
    size_t i = (size_t)blockIdx.x * blockDim.x + threadIdx.x;
    if (i >= (size_t)NN * Z1W) return;
    float v = r1[i] + b1[i & (Z1W - 1)];
    r1[i] = v > 0.f ? v : 0.f;
}

// ---------------- K7: init layer-2 accumulators (r2 aliases dead z1) ----------------
__global__ void init2_kernel(float* __restrict__ r2, float* __restrict__ m2, float* __restrict__ den2) {
    size_t i = (size_t)blockIdx.x * blockDim.x + threadIdx.x;
    size_t stride = (size_t)gridDim.x * blockDim.x;
    for (size_t k = i; k < (size_t)NN * Z2W; k += stride) r2[k] = 0.f;
    for (size_t k = i; k < (size_t)NN * 4; k += stride) { m2[k] = -1e30f; den2[k] = 0.f; }
}

// ---------------- K8: z2 = r1(200000x64) @ W2(64x60) via V_WMMA_F32_16X16X4_F32 ----------------
// A 16x4 f32 layout: lanes 0-15 -> K={0,1} by VGPR, lanes 16-31 -> K={2,3}.
// B 4x16 f32 layout: mirrored (lane half selects K pair, VGPR selects K within pair).
// C/D 16x16 f32: VGPR r holds M = r + (lane>=16 ? 8 : 0), N = lane%16.
// Tiles staged to LDS via GLOBAL_LOAD_ASYNC_TO_LDS_B128 (ASYNCcnt) when available.
// Pitch 68 floats: 272B rows keep B128 16B alignment; bank step = 4 -> 16 rows
// span all 64 banks, and the two lane-halves (kb parity) hit disjoint banks.
__global__ __launch_bounds__(256) void gemm2_kernel(const float* __restrict__ r1,
                                                    const float* __restrict__ W2,
                                                    float* __restrict__ z2) {
    __shared__ float sA[128][68];   // 128 rows x 64 K
    __shared__ float sB[64][68];    // 64 K x 64 cols (60 used, padded)
    const int row0 = blockIdx.x * 128;
    const int tid = threadIdx.x;

#if HAVE_ASYNC_LDS
    // A strip: 128 rows x 16 float4 each. Rows past GROWS read in-workspace
    // garbage; those output rows are masked on store.
    for (int idx = tid; idx < 128 * 16; idx += 256) {
        int r = idx >> 4, q = idx & 15;
        const float* g = r1 + (size_t)(row0 + r) * HID + q * 4;
        __builtin_amdgcn_global_load_async_to_lds_b128(AS_GLOBAL(g), AS_LDS(&sA[r][q * 4]), 0, 0);
    }
    // B tile: 64 rows x 15 float4 (60 floats per row, 240B rows are 16B aligned)
    for (int idx = tid; idx < 64 * 15; idx += 256) {
        int k = idx / 15, q = idx - k * 15;
        const float* g = W2 + k * 60 + q * 4;
        __builtin_amdgcn_global_load_async_to_lds_b128(AS_GLOBAL(g), AS_LDS(&sB[k][q * 4]), 0, 0);
    }
    if (tid < 64) { sB[tid][60] = 0.f; sB[tid][61] = 0.f; sB[tid][62] = 0.f; sB[tid][63] = 0.f; }
#if __has_builtin(__builtin_amdgcn_s_wait_asynccnt)
    __builtin_amdgcn_s_wait_asynccnt(0);
#else
    asm volatile("s_wait_asynccnt 0x0" ::: "memory");
#endif
#else
    for (int idx = tid; idx < 128 * 64; idx += 256) {
        int r = idx >> 6, c = idx & 63;
        int gr = row0 + r;
        sA[r][c] = (gr < GROWS) ? r1[(size_t)gr * HID + c] : 0.f;
    }
    for (int idx = tid; idx < 64 * 64; idx += 256) {
        int k = idx >> 6, c = idx & 63;
        sB[k][c] = (c < 60) ? W2[k * 60 + c] : 0.f;
    }
#endif
    __syncthreads();

    const int wave = tid >> 5;
    const int lane = tid & 31;
    const int lrow = (wave << 4) + (lane & 15);
    const int khi  = (lane >> 4) << 1;           // 0 or 2
    const int coll = lane & 15;

    for (int ct = 0; ct < 4; ++ct) {
        v8f acc = {};
        const int col = (ct << 4) + coll;
        #pragma unroll
        for (int ks = 0; ks < 16; ++ks) {
            const int kb = (ks << 2) + khi;
            v2f a; a.x = sA[lrow][kb];  a.y = sA[lrow][kb + 1];
            v2f b; b.x = sB[kb][col];   b.y = sB[kb + 1][col];
            acc = __builtin_amdgcn_wmma_f32_16x16x4_f32(false, a, false, b, (short)0, acc,
                                                        false, false);
        }
        const int growBase = row0 + (wave << 4) + ((lane >> 4) << 3);
        if (col < 60) {
            #pragma unroll
            for (int r = 0; r < 8; ++r) {
                int grow = growBase + r;
                if (grow < GROWS) z2[(size_t)grow * 60 + col] = acc[r];
            }
        }
    }
}

// ---------------- K9: el2/er2 per node (4 channels) ----------------
__global__ __launch_bounds__(128) void attn2_kernel(const float* __restrict__ z2,
                                                    const float* __restrict__ al2,
                                                    const float* __restrict__ ar2,
                                                    float* __restrict__ el2,
                                                    float* __restrict__ er2) {
    int n = blockIdx.x, t = threadIdx.x;
    __shared__ float sl[128], sr[128];
    float zl = 0.f, zr = 0.f;
    if (t < Z2W) {
        float z = z2[(size_t)n * Z2W + t];
        int c = t / EMB, e = t - c * EMB;   // c = h1*2 + h2
        int h2 = c & 1;
        zl = z * al2[h2 * EMB + e];
        zr = z * ar2[h2 * EMB + e];
    }
    sl[t] = zl; sr[t] = zr;
    __syncthreads();
    if (t < 4) {
        float a = 0.f, b = 0.f;
        for (int e = 0; e < EMB; ++e) { a += sl[t * EMB + e]; b += sr[t * EMB + e]; }
        el2[n * 4 + t] = a; er2[n * 4 + t] = b;
    }
}

// ---------------- K10/K11: layer-2 edge softmax ----------------
__global__ void emax2_kernel(const int* __restrict__ src, const int* __restrict__ dst,
                             const float* __restrict__ el, const float* __restrict__ er,
                             float* __restrict__ m) {
    int i = blockIdx.x * blockDim.x + threadIdx.x;
    if (i >= EE) return;
    int s = src[i], d = dst[i];
    for (int c = 0; c < 4; ++c) atomicMaxF(&m[d * 4 + c], lrelu(el[s * 4 + c] + er[d * 4 + c]));
}

__global__ void eden2_kernel(const int* __restrict__ src, const int* __restrict__ dst,
                             const float* __restrict__ el, const float* __restrict__ er,
                             const float* __restrict__ m, float* __restrict__ den) {
    int i = blockIdx.x * blockDim.x + threadIdx.x;
    if (i >= EE) return;
    int s = src[i], d = dst[i];
    for (int c = 0; c < 4; ++c) {
        float e = lrelu(el[s * 4 + c] + er[d * 4 + c]);
        atomicAdd(&den[d * 4 + c], __expf(e - m[d * 4 + c]));
    }
}

// ---------------- K12: layer-2 weighted scatter (wave32 per edge) ----------------
__global__ __launch_bounds__(256) void agg2_kernel(const int* __restrict__ src, const int* __restrict__ dst,
                                                   const float* __restrict__ el, const float* __restrict__ er,
                                                   const float* __restrict__ m, const float* __restrict__ den,
                                                   const float* __restrict__ z2, float* __restrict__ r2) {
    int eid = (blockIdx.x * blockDim.x + threadIdx.x) >> 5;
    int lane = threadIdx.x & 31;
    if (eid >= EE) return;
    int s = src[eid], d = dst[eid];
    int j0 = lane * 4;
    const float* zrow = z2 + (size_t)s * Z2W + j0;
    if (j0 < Z2W) __builtin_prefetch(zrow, 0, 1);
    float my = 0.f;
    if (lane < 4) {
        float e = lrelu(el[s * 4 + lane] + er[d * 4 + lane]);
        my = __expf(e - m[d * 4 + lane]) / den[d * 4 + lane];
    }
    float a0 = __shfl(my, 0, 32), a1 = __shfl(my, 1, 32);
    float a2 = __shfl(my, 2, 32), a3 = __shfl(my, 3, 32);
    float al4[4] = {a0, a1, a2, a3};
    if (j0 < Z2W) {
        const float4 zv = *(const float4*)zrow;
        float zz[4] = {zv.x, zv.y, zv.z, zv.w};
        float* out = r2 + (size_t)d * Z2W;
        #pragma unroll
        for (int u = 0; u < 4; ++u) {
            int j = j0 + u;
            if (j < Z2W) atomicAdd(out + j, al4[j / EMB] * zz[u]);
        }
    }
}

// ---------------- K13: mean over nodes of channel (0,0) ----------------
__global__ __launch_bounds__(256) void mean_kernel(const float* __restrict__ r2, float* __restrict__ a00) {
    int e = threadIdx.x & 31;
    int rowInBlk = threadIdx.x >> 5;   // 8 node-rows per block
    if (e >= EMB) return;
    float sum = 0.f;
    for (int n = blockIdx.x * 8 + rowInBlk; n < NN; n += gridDim.x * 8)
        sum += r2[(size_t)n * Z2W + e];
    atomicAdd(a00 + e, sum);
}

// ---------------- K14: assemble h, W_lin4 @ h, log_softmax ----------------
__global__ __launch_bounds__(128) void final_kernel(const float* __restrict__ a00,
                                                    const float* __restrict__ h_img,
                                                    const float* __restrict__ vocab,
                                                    const float* __restrict__ b2,
                                                    const float* __restrict__ W_lin4,
                                                    float* __restrict__ out) {
    __shared__ float h[2 * EMB + VOCAB];
    __shared__ float logit[2];
    int t = threadIdx.x;
    if (t < EMB)                       h[t] = a00[t] / (float)NN + b2[t];
    else if (t < 2 * EMB)              h[t] = h_img[t - EMB];
    else if (t < 2 * EMB + VOCAB)      h[t] = vocab[t - 2 * EMB];
    __syncthreads();
    if (t < 2) {
        float acc = 0.f;
        for (int j = 0; j < 2 * EMB + VOCAB; ++j) acc += W_lin4[t * (2 * EMB + VOCAB) + j] * h[j];
        logit[t] = acc;
    }
    __syncthreads();
    if (t == 0) {
        float mx = fmaxf(logit[0], logit[1]);
        float lse = mx + logf(expf(logit[0] - mx) + expf(logit[1] - mx));
        out[0] = logit[0] - lse;
        out[1] = logit[1] - lse;
    }
}

// ---------------- host launcher ----------------
extern "C" void kernel_launch(void* const* d_in, const int* in_sizes, int n_in,
                              void* d_out, int out_size, void* d_ws, size_t ws_size,
                              hipStream_t stream) {
    const float* x      = (const float*)d_in[0];
    const float* feat   = (const float*)d_in[1];
    const float* vocab  = (const float*)d_in[2];
    const int*   src    = (const int*)d_in[3];
    const int*   dst    = (const int*)d_in[4];
    const float* W_lin1 = (const float*)d_in[5];
    const float* w2c    = (const float*)d_in[6];
    const float* w3c    = (const float*)d_in[7];
    const float* W_lin4 = (const float*)d_in[8];
    const float* W1     = (const float*)d_in[9];
    const float* al1    = (const float*)d_in[10];
    const float* ar1    = (const float*)d_in[11];
    const float* b1     = (const float*)d_in[12];
    const float* W2     = (const float*)d_in[13];
    const float* al2    = (const float*)d_in[14];
    const float* ar2    = (const float*)d_in[15];
    const float* b2     = (const float*)d_in[16];
    float* out = (float*)d_out;
    float* ws  = (float*)d_ws;

    float* z1   = ws + OFF_Z1;
    float* r1   = ws + OFF_R1;
    float* z2   = ws + OFF_Z2;
    float* r2   = ws + OFF_R2;    // aliases z1 (z1 dead by then)
    float* el1  = ws + OFF_EL1;
    float* er1  = ws + OFF_ER1;
    float* m1   = ws + OFF_M1;
    float* den1 = ws + OFF_DEN1;
    float* el2  = ws + OFF_EL2;
    float* er2  = ws + OFF_ER2;
    float* m2   = ws + OFF_M2;
    float* den2 = ws + OFF_DEN2;
    float* a00  = ws + OFF_A00;
    float* himg = ws + OFF_HIMG;

    img_kernel<<<1, 64, 0, stream>>>(x, W_lin1, w2c, w3c, himg);
    init1_kernel<<<2048, 256, 0, stream>>>(r1, m1, den1, a00);
    z1_kernel<<<NN, 128, 0, stream>>>(feat, W1, al1, ar1, z1, el1, er1);
    emax1_kernel<<<(EE + 255) / 256, 256, 0, stream>>>(src, dst, el1, er1, m1);
    eden1_kernel<<<(EE + 255) / 256, 256, 0, stream>>>(src, dst, el1, er1, m1, den1);
    agg1_kernel<<<(EE * 32 + 255) / 256, 256, 0, stream>>>(src, dst, el1, er1, m1, den1, z1, r1);
    bias1_kernel<<<(NN * Z1W + 255) / 256, 256, 0, stream>>>(r1, b1);
    init2_kernel<<<2048, 256, 0, stream>>>(r2, m2, den2);
    gemm2_kernel<<<(GROWS + 127) / 128, 256, 0, stream>>>(r1, W2, z2);
    attn2_kernel<<<NN, 128, 0, stream>>>(z2, al2, ar2, el2, er2);
    emax2_kernel<<<(EE + 255) / 256, 256, 0, stream>>>(src, dst, el2, er2, m2);
    eden2_kernel<<<(EE + 255) / 256, 256, 0, stream>>>(src, dst, el2, er2, m2, den2);
    agg2_kernel<<<(EE * 32 + 255) / 256, 256, 0, stream>>>(src, dst, el2, er2, m2, den2, z2, r2);
    mean_kernel<<<512, 256, 0, stream>>>(r2, a00);
    final_kernel<<<1, 128, 0, stream>>>(a00, himg, vocab, b2, W_lin4, out);
}